// VICReg_22084721836287
// MI455X (gfx1250) — compile-verified
//
#include <hip/hip_runtime.h>
#include <math.h>

// Problem constants (from the reference)
#define NN 2048
#define DD 2048
#define DT 512
#define VAR_EPS 1e-4f

// Gram tiling
#define TILE 128
#define KT 32
#define NT (DD / TILE)            // 16 tile rows/cols
#define NTRI (NT * (NT + 1) / 2)  // 136 upper-triangle tiles
#define LSTRIDE (TILE + 4)        // LDS pad (528 B rows, 16B aligned)

typedef __attribute__((ext_vector_type(2))) float v2f;
typedef __attribute__((ext_vector_type(8))) float v8f;

// ---------------------------------------------------------------------------
// 1) Per-column stats of x and y: means (for Gram centering) + relu(1-std)
//    partial sums (for std_loss). 8 blocks x 256 threads, 1 column/thread.
// ---------------------------------------------------------------------------
__global__ void vicreg_colstats(const float* __restrict__ x,
                                const float* __restrict__ y,
                                float* __restrict__ mean_x,
                                float* __restrict__ mean_y,
                                float* __restrict__ relux,
                                float* __restrict__ reluy) {
  int col = blockIdx.x * 256 + threadIdx.x;
  float sx = 0.f, sxx = 0.f, sy = 0.f, syy = 0.f;
  for (int i = 0; i < NN; ++i) {
    float vx = x[(size_t)i * DD + col];
    float vy = y[(size_t)i * DD + col];
    sx += vx; sxx += vx * vx;
    sy += vy; syy += vy * vy;
  }
  float mx = sx * (1.0f / NN), my = sy * (1.0f / NN);
  mean_x[col] = mx;
  mean_y[col] = my;
  float varx = (sxx - sx * mx) * (1.0f / (NN - 1));  // ddof=1
  float vary = (syy - sy * my) * (1.0f / (NN - 1));
  float rx = fmaxf(0.f, 1.f - sqrtf(varx + VAR_EPS));
  float ry = fmaxf(0.f, 1.f - sqrtf(vary + VAR_EPS));

  __shared__ float red[256];
  red[threadIdx.x] = rx; __syncthreads();
  for (int s = 128; s > 0; s >>= 1) {
    if (threadIdx.x < s) red[threadIdx.x] += red[threadIdx.x + s];
    __syncthreads();
  }
  if (threadIdx.x == 0) relux[blockIdx.x] = red[0];
  __syncthreads();
  red[threadIdx.x] = ry; __syncthreads();
  for (int s = 128; s > 0; s >>= 1) {
    if (threadIdx.x < s) red[threadIdx.x] += red[threadIdx.x + s];
    __syncthreads();
  }
  if (threadIdx.x == 0) reluy[blockIdx.x] = red[0];
}

// ---------------------------------------------------------------------------
// 2) Only row 0 of the similarity matrix is used. Per target-row j:
//    dot0[j] = sum_d |t0[d]|*|tj[d]|,  nsq[j] = ||t_j||^2. One wave32 per j.
// ---------------------------------------------------------------------------
__global__ void vicreg_tdot(const float* __restrict__ t,
                            float* __restrict__ dot0,
                            float* __restrict__ nsq) {
  int wave = threadIdx.x >> 5, lane = threadIdx.x & 31;
  int j = blockIdx.x * 8 + wave;
  float d0 = 0.f, ns = 0.f;
  for (int d = lane; d < DT; d += 32) {
    float tj = t[(size_t)j * DT + d];
    float t0 = t[d];
    d0 += fabsf(t0) * fabsf(tj);
    ns += tj * tj;
  }
  for (int off = 16; off > 0; off >>= 1) {
    d0 += __shfl_down(d0, off, 32);
    ns += __shfl_down(ns, off, 32);
  }
  if (lane == 0) { dot0[j] = d0; nsq[j] = ns; }
}

__global__ void vicreg_srow(const float* __restrict__ dot0,
                            const float* __restrict__ nsq,
                            float* __restrict__ s) {
  int j = blockIdx.x * 256 + threadIdx.x;
  float n0 = sqrtf(nsq[0]);
  float nj = sqrtf(nsq[j]);
  s[j] = sqrtf(dot0[j]) / fmaxf(n0, nj);
}

// ---------------------------------------------------------------------------
// 3) repr_loss partials: sum_{i,d} s[d]*(x-y)^2, fixed-order block reduction.
// ---------------------------------------------------------------------------
__global__ void vicreg_repr(const float* __restrict__ x,
                            const float* __restrict__ y,
                            const float* __restrict__ s,
                            float* __restrict__ part) {
  size_t total = (size_t)NN * DD;
  float acc = 0.f;
  for (size_t idx = (size_t)blockIdx.x * 256 + threadIdx.x; idx < total;
       idx += (size_t)gridDim.x * 256) {
    int d = (int)(idx & (DD - 1));
    float dv = x[idx] - y[idx];
    acc += s[d] * dv * dv;
  }
  __shared__ float red[256];
  red[threadIdx.x] = acc; __syncthreads();
  for (int st = 128; st > 0; st >>= 1) {
    if (threadIdx.x < st) red[threadIdx.x] += red[threadIdx.x + st];
    __syncthreads();
  }
  if (threadIdx.x == 0) part[blockIdx.x] = red[0];
}

// ---------------------------------------------------------------------------
// 4) Gram kernel: R = X^T X via V_WMMA_F32_16X16X4_F32.
//    - K staged through LDS in 32-deep panels, DOUBLE-BUFFERED with
//      GLOBAL_LOAD_ASYNC_TO_LDS_B128 (ASYNCcnt) so the next panel's
//      global->LDS DMA overlaps the current panel's 64 WMMAs.
//    - Epilogue applies G = R - N*mu_a*mu_b and accumulates sum(G^2)
//      (x2 for off-diagonal tiles) and diag(G)^2; G never materialized.
//    - Upper-triangle 128x128 tiles only (symmetry). 256 thr = 8 waves.
// ---------------------------------------------------------------------------
__global__ void __launch_bounds__(256)
vicreg_gram(const float* __restrict__ x, const float* __restrict__ y,
            const float* __restrict__ mean_x, const float* __restrict__ mean_y,
            float* __restrict__ cov_ss, float* __restrict__ cov_dg) {
  __shared__ __align__(16) float tA2[2][KT * LSTRIDE];
  __shared__ __align__(16) float tB2[2][KT * LSTRIDE];
  __shared__ float red[256];

  const float* src = (blockIdx.y == 0) ? x : y;
  const float* mu  = (blockIdx.y == 0) ? mean_x : mean_y;

  // map linear triangle index -> (ta, tb), ta <= tb
  int rem = blockIdx.x, ta = 0;
  while (rem >= (NT - ta)) { rem -= (NT - ta); ++ta; }
  int tb = ta + rem;
  int abase = ta * TILE, bbase = tb * TILE;

  int tid = threadIdx.x;
  int wave = tid >> 5, lane = tid & 31;
  int wr = wave & 3;        // a-direction: 4 waves x 32 rows
  int wc = wave >> 2;       // b-direction: 2 waves x 64 cols
  int li = lane & 15, half = lane >> 4;

  // Per-thread slot for cooperative async panel loads: 4 x B128 per panel.
  int lrow[4], lc4[4];
#pragma unroll
  for (int q = 0; q < 4; ++q) {
    int f = tid + q * 256;          // 0..1023 float4 slots
    lrow[q] = f >> 5;               // 0..31
    lc4[q] = (f & 31) * 4;          // 0..124
  }

  // Issue one stage's async global->LDS copies (EXEC all-ones).
  auto issue_stage = [&](int k0, int b) {
#pragma unroll
    for (int q = 0; q < 4; ++q) {
      const float* ga = src + (size_t)(k0 + lrow[q]) * DD + abase + lc4[q];
      unsigned la =
          (unsigned)(unsigned long long)(const void*)&tA2[b][lrow[q] * LSTRIDE + lc4[q]];
      asm volatile("global_load_async_to_lds_b128 %0, %1, off"
                   :: "v"(la), "v"(ga) : "memory");
      const float* gb = src + (size_t)(k0 + lrow[q]) * DD + bbase + lc4[q];
      unsigned lb =
          (unsigned)(unsigned long long)(const void*)&tB2[b][lrow[q] * LSTRIDE + lc4[q]];
      asm volatile("global_load_async_to_lds_b128 %0, %1, off"
                   :: "v"(lb), "v"(gb) : "memory");
    }
  };

  v8f acc[2][4] = {};       // 32x64 per-wave accumulator (8 x 16x16 tiles)

  // Prologue: fill buffer 0.
  issue_stage(0, 0);
  asm volatile("s_wait_asynccnt 0x0" ::: "memory");
  __syncthreads();

  int buf = 0;
  for (int k0 = 0; k0 < NN; k0 += KT, buf ^= 1) {
    // Prefetch next K-panel into the other buffer (overlaps the WMMAs).
    if (k0 + KT < NN) issue_stage(k0 + KT, buf ^ 1);

    const float* tA = tA2[buf];
    const float* tB = tB2[buf];
#pragma unroll
    for (int kk = 0; kk < KT / 4; ++kk) {
      // fp32 16x16x4 fragment layout: lane (li,half) holds K=2*half,2*half+1
      int r0 = kk * 4 + 2 * half;
      v2f aF[2], bF[4];
#pragma unroll
      for (int i = 0; i < 2; ++i) {
        int ca = wr * 32 + i * 16 + li;
        aF[i].x = tA[r0 * LSTRIDE + ca];
        aF[i].y = tA[(r0 + 1) * LSTRIDE + ca];
      }
#pragma unroll
      for (int j = 0; j < 4; ++j) {
        int cb = wc * 64 + j * 16 + li;
        bF[j].x = tB[r0 * LSTRIDE + cb];
        bF[j].y = tB[(r0 + 1) * LSTRIDE + cb];
      }
#pragma unroll
      for (int i = 0; i < 2; ++i)
#pragma unroll
        for (int j = 0; j < 4; ++j)
          acc[i][j] = __builtin_amdgcn_wmma_f32_16x16x4_f32(
              false, aF[i], false, bF[j], (short)0, acc[i][j], false, false);
    }

    // All waves done reading buf; all async loads into buf^1 complete.
    asm volatile("s_wait_asynccnt 0x0" ::: "memory");
    __syncthreads();
  }

  // Epilogue: center (G = R - N*mu_a*mu_b), square, reduce.
  float ssum = 0.f, dsum = 0.f;
  float w2 = (ta == tb) ? 1.0f : 2.0f;
#pragma unroll
  for (int i = 0; i < 2; ++i) {
#pragma unroll
    for (int j = 0; j < 4; ++j) {
      int bcol = bbase + wc * 64 + j * 16 + li;
      float mb = mu[bcol];
#pragma unroll
      for (int r = 0; r < 8; ++r) {
        int arow = abase + wr * 32 + i * 16 + r + 8 * half;  // C/D layout
        float g = acc[i][j][r] - (float)NN * mu[arow] * mb;
        ssum += w2 * g * g;
        if (ta == tb && arow == bcol) dsum += g * g;
      }
    }
  }
  red[tid] = ssum; __syncthreads();
  for (int st = 128; st > 0; st >>= 1) {
    if (tid < st) red[tid] += red[tid + st];
    __syncthreads();
  }
  if (tid == 0) cov_ss[blockIdx.y * NTRI + blockIdx.x] = red[0];
  __syncthreads();
  red[tid] = dsum; __syncthreads();
  for (int st = 128; st > 0; st >>= 1) {
    if (tid < st) red[tid] += red[tid + st];
    __syncthreads();
  }
  if (tid == 0) cov_dg[blockIdx.y * NTRI + blockIdx.x] = red[0];
}

// ---------------------------------------------------------------------------
// 5) Final deterministic combine (single thread, fixed order, trivial work).
// ---------------------------------------------------------------------------
__global__ void vicreg_final(const float* __restrict__ relux,
                             const float* __restrict__ reluy,
                             const float* __restrict__ reprp,
                             const float* __restrict__ cov_ss,
                             const float* __restrict__ cov_dg,
                             float* __restrict__ out) {
  float rx = 0.f, ry = 0.f;
  for (int i = 0; i < 8; ++i) { rx += relux[i]; ry += reluy[i]; }
  float std_loss = rx / (2.0f * DD) + ry / (2.0f * DD);
  float rp = 0.f;
  for (int i = 0; i < 128; ++i) rp += reprp[i];
  float repr_loss = rp / ((float)NN * (float)DD);
  float ssx = 0.f, dgx = 0.f, ssy = 0.f, dgy = 0.f;
  for (int i = 0; i < NTRI; ++i) {
    ssx += cov_ss[i];        dgx += cov_dg[i];
    ssy += cov_ss[NTRI + i]; dgy += cov_dg[NTRI + i];
  }
  float inv = 1.0f / ((float)(NN - 1) * (float)(NN - 1));
  float cov_loss = ((ssx - dgx) * inv + (ssy - dgy) * inv) / (float)DD;
  out[0] = repr_loss + std_loss + cov_loss;
}

// ---------------------------------------------------------------------------
extern "C" void kernel_launch(void* const* d_in, const int* in_sizes, int n_in,
                              void* d_out, int out_size, void* d_ws,
                              size_t ws_size, hipStream_t stream) {
  const float* x = (const float*)d_in[0];
  const float* y = (const float*)d_in[1];
  const float* t = (const float*)d_in[2];
  float* out = (float*)d_out;

  float* ws = (float*)d_ws;
  float* mean_x = ws + 0;      // 2048
  float* mean_y = ws + 2048;   // 2048
  float* s      = ws + 4096;   // 2048
  float* dot0   = ws + 6144;   // 2048
  float* nsq    = ws + 8192;   // 2048
  float* relux  = ws + 10240;  // 8
  float* reluy  = ws + 10248;  // 8
  float* reprp  = ws + 10256;  // 128
  float* cov_ss = ws + 10384;  // 2*136
  float* cov_dg = ws + 10656;  // 2*136  (total ~43 KB)

  vicreg_colstats<<<8, 256, 0, stream>>>(x, y, mean_x, mean_y, relux, reluy);
  vicreg_tdot<<<256, 256, 0, stream>>>(t, dot0, nsq);
  vicreg_srow<<<8, 256, 0, stream>>>(dot0, nsq, s);
  vicreg_repr<<<128, 256, 0, stream>>>(x, y, s, reprp);
  vicreg_gram<<<dim3(NTRI, 2), 256, 0, stream>>>(x, y, mean_x, mean_y,
                                                 cov_ss, cov_dg);
  vicreg_final<<<1, 1, 0, stream>>>(relux, reluy, reprp, cov_ss, cov_dg, out);
}